// SSMKernel_8203387535800
// MI455X (gfx1250) — compile-verified
//
#include <hip/hip_runtime.h>

// ---------------------------------------------------------------------------
// Chunked SSD reformulation of the diagonal SSM scan for MI455X (gfx1250).
//   B=8, L=4096, D=1024, N=16 states. Chunk T=16, 16 chunks/group, 16 groups.
// Per wave (one batch b, one channel d), per 256-timestep group:
//   GEMM1: Yintra = TriK(16x16) @ Xc(16x16)        (V_WMMA_F32_16X16X4_F32 x4)
//   GEMM2: S      = Wt(16x16)   @ Xc(16x16)        (x4)
//   scan : h <- dA^16 * h + S[:,c]  (16 serial steps, lanes 0..15)
//   GEMM3: Y += CdA(16x16) @ H(16x16)              (x4)
//   Y += D * x ; coalesced store.
// Reads x once, writes out once -> ~268 MB -> HBM-roofline bound (~11.5us).
// ---------------------------------------------------------------------------

#define DM   1024
#define NS   16
#define LSEQ 4096
#define WSC  800   // floats of workspace per channel
// ws float offsets per channel: TriK @0 (256), Wt @256 (256), CdA @512 (256),
//                               dAT @768 (16), D @784 (1)

typedef __attribute__((ext_vector_type(2))) float v2f;
typedef __attribute__((ext_vector_type(8))) float v8f;

__device__ __forceinline__ v8f wmma4(v2f a, v2f b, v8f c) {
  // D = A(16x4) * B(4x16) + C ; fp32 WMMA, 8-arg form:
  // (neg_a, A, neg_b, B, c_mod, C, reuse_a, reuse_b)
  return __builtin_amdgcn_wmma_f32_16x16x4_f32(false, a, false, b, (short)0, c,
                                               false, false);
}

// ---------------- per-channel parameter tables --------------------------------
__global__ __launch_bounds__(256) void ssm_precompute(
    const float* __restrict__ A_log, const float* __restrict__ Bm,
    const float* __restrict__ C, const float* __restrict__ Dv,
    const float* __restrict__ dt, float* __restrict__ ws) {
  int d = blockIdx.x * blockDim.x + threadIdx.x;
  if (d >= DM) return;
  float dtv   = dt[d];
  float dt_sp = (dtv > 20.f) ? dtv : log1pf(expf(dtv));  // softplus
  float* wsd  = ws + (size_t)d * WSC;

  float Kk[16];
#pragma unroll
  for (int i = 0; i < 16; ++i) Kk[i] = 0.f;
  float dATl[16];

#pragma unroll
  for (int n = 0; n < NS; ++n) {
    float a  = -expf(A_log[d * NS + n]);
    float dA = expf(dt_sp * a);
    float dB = dt_sp * Bm[d * NS + n];
    float c  = C[d * NS + n];
    float pw[17];
    float p = 1.f;
#pragma unroll
    for (int i = 0; i < 17; ++i) { pw[i] = p; p *= dA; }
#pragma unroll
    for (int t = 0; t < 16; ++t) Kk[t] += c * pw[t] * dB;          // kernel K
#pragma unroll
    for (int t = 0; t < 16; ++t) wsd[512 + t * 16 + n] = c * pw[t + 1]; // CdA
#pragma unroll
    for (int s = 0; s < 16; ++s) wsd[256 + n * 16 + s] = pw[15 - s] * dB; // Wt
    dATl[n] = pw[16];
  }
#pragma unroll
  for (int t = 0; t < 16; ++t)
#pragma unroll
    for (int s = 0; s < 16; ++s)
      wsd[t * 16 + s] = (t >= s) ? Kk[t - s] : 0.f;                // TriK
#pragma unroll
  for (int n = 0; n < 16; ++n) wsd[768 + n] = dATl[n];
  wsd[784] = Dv[d];
}

// ---------------- main fused scan kernel --------------------------------------
__global__ __launch_bounds__(512) void ssm_scan(
    const float* __restrict__ x, const float* __restrict__ ws,
    float* __restrict__ out) {
  __shared__ float Xs[256 * 17];          // 256 timesteps x 16 channels, pitch 17
  __shared__ float SHs[16][2][16 * 17];   // per-wave S and H tiles, pitch 17

  const int dTile = blockIdx.x;   // 0..63
  const int b     = blockIdx.y;   // 0..7
  const int tid   = threadIdx.x;
  const int wave  = tid >> 5;     // channel within tile
  const int lane  = tid & 31;
  const int d     = dTile * 16 + wave;
  const int m     = lane & 15;          // A-op row M / B-op & acc column N
  const int koff  = (lane >> 4) * 2;    // K sub-offset (VGPR0 holds K 0|2)
  const int hi8   = (lane >> 4) * 8;    // acc M offset for hi half-wave

  // per-channel constant matrices in WMMA A layout (4 K-slices x 2 floats)
  const float* wsd = ws + (size_t)d * WSC;
  float triA[4][2], wtA[4][2], cdaA[4][2];
#pragma unroll
  for (int k = 0; k < 4; ++k)
#pragma unroll
    for (int j = 0; j < 2; ++j) {
      int kk = 4 * k + koff + j;
      triA[k][j] = wsd[      m * 16 + kk];
      wtA [k][j] = wsd[256 + m * 16 + kk];
      cdaA[k][j] = wsd[512 + m * 16 + kk];
    }
  const float dAT = (lane < 16) ? wsd[768 + lane] : 0.f;
  const float Dd  = wsd[784];

  const float* xb = x   + ((size_t)b * LSEQ) * DM + (size_t)dTile * 16;
  float*       ob = out + ((size_t)b * LSEQ) * DM + (size_t)dTile * 16;

  const int q  = tid & 3;    // float4 column within the 16-channel row
  const int r0 = tid >> 2;   // staging row 0..127

  float* Sl = &SHs[wave][0][0];
  float* Hl = &SHs[wave][1][0];

  float hcarry = 0.f;        // lane n<16 holds state n for this (b,d)

  for (int g = 0; g < 16; ++g) {
    __syncthreads();  // previous group's cooperative readers are done with Xs

    // ---- coalesced stage: Xs[t][ch] = x[b, g*256+t, dTile*16+ch] ----
#pragma unroll
    for (int i = 0; i < 2; ++i) {
      int r = r0 + i * 128;
      const float4 v = *(const float4*)(xb + (size_t)(g * 256 + r) * DM + q * 4);
      Xs[r * 17 + q * 4 + 0] = v.x;
      Xs[r * 17 + q * 4 + 1] = v.y;
      Xs[r * 17 + q * 4 + 2] = v.z;
      Xs[r * 17 + q * 4 + 3] = v.w;
    }
    __syncthreads();

    // ---- B operand Xc[s][c], c = m, s = K index; shared by GEMM1/GEMM2 ----
    float bX[4][2];
#pragma unroll
    for (int k = 0; k < 4; ++k)
#pragma unroll
      for (int j = 0; j < 2; ++j) {
        int s = 4 * k + koff + j;
        bX[k][j] = Xs[(m * 16 + s) * 17 + wave];
      }

    v8f acc = {0.f, 0.f, 0.f, 0.f, 0.f, 0.f, 0.f, 0.f};
    v8f sS  = {0.f, 0.f, 0.f, 0.f, 0.f, 0.f, 0.f, 0.f};
#pragma unroll
    for (int k = 0; k < 4; ++k) {
      v2f a1 = {triA[k][0], triA[k][1]};
      v2f a2 = {wtA [k][0], wtA [k][1]};
      v2f bb = {bX  [k][0], bX  [k][1]};
      acc = wmma4(a1, bb, acc);   // Yintra accumulation
      sS  = wmma4(a2, bb, sS);    // chunk-state contributions
    }

    // ---- spill S (C-layout: M = v+hi8 = n, N = m = chunk) ----
#pragma unroll
    for (int v = 0; v < 8; ++v) Sl[(v + hi8) * 17 + m] = sS[v];

    // ---- serial inter-chunk recurrence: lane n, 16 chunks ----
    if (lane < 16) {
      float h = hcarry;
#pragma unroll
      for (int c = 0; c < 16; ++c) {
        Hl[lane * 17 + c] = h;               // state entering chunk c
        h = dAT * h + Sl[lane * 17 + c];
      }
      hcarry = h;
    }

    // ---- GEMM3: acc += CdA[t][n] @ H[n][c] ----
    float bH[4][2];
#pragma unroll
    for (int k = 0; k < 4; ++k)
#pragma unroll
      for (int j = 0; j < 2; ++j) {
        int s = 4 * k + koff + j;
        bH[k][j] = Hl[s * 17 + m];
      }
#pragma unroll
    for (int k = 0; k < 4; ++k) {
      v2f a3 = {cdaA[k][0], cdaA[k][1]};
      v2f bb = {bH  [k][0], bH  [k][1]};
      acc = wmma4(a3, bb, acc);
    }

    // ---- skip connection: Y += D * x ----
#pragma unroll
    for (int v = 0; v < 8; ++v) {
      int tg = m * 16 + v + hi8;             // group-local timestep
      acc[v] += Dd * Xs[tg * 17 + wave];
    }

    __syncthreads();  // all waves done reading Xs
    // ---- restage Y into Xs, then coalesced store ----
#pragma unroll
    for (int v = 0; v < 8; ++v) {
      int tg = m * 16 + v + hi8;
      Xs[tg * 17 + wave] = acc[v];
    }
    __syncthreads();

#pragma unroll
    for (int i = 0; i < 2; ++i) {
      int r = r0 + i * 128;
      float4 o;
      o.x = Xs[r * 17 + q * 4 + 0];
      o.y = Xs[r * 17 + q * 4 + 1];
      o.z = Xs[r * 17 + q * 4 + 2];
      o.w = Xs[r * 17 + q * 4 + 3];
      *(float4*)(ob + (size_t)(g * 256 + r) * DM + q * 4) = o;
    }
  }
}

// ---------------- launcher ----------------------------------------------------
extern "C" void kernel_launch(void* const* d_in, const int* in_sizes, int n_in,
                              void* d_out, int out_size, void* d_ws,
                              size_t ws_size, hipStream_t stream) {
  (void)in_sizes; (void)n_in; (void)out_size; (void)ws_size;
  const float* x     = (const float*)d_in[0];
  const float* A_log = (const float*)d_in[1];
  const float* Bm    = (const float*)d_in[2];
  const float* C     = (const float*)d_in[3];
  const float* Dv    = (const float*)d_in[4];
  const float* dt    = (const float*)d_in[5];
  float* ws  = (float*)d_ws;   // needs 1024*800*4 = 3.2 MB
  float* out = (float*)d_out;

  ssm_precompute<<<dim3(4), dim3(256), 0, stream>>>(A_log, Bm, C, Dv, dt, ws);
  ssm_scan<<<dim3(64, 8), dim3(512), 0, stream>>>(x, ws, out);
}